// GCNPolicyNetwork_7670811590824
// MI455X (gfx1250) — compile-verified
//
#include <hip/hip_runtime.h>
#include <hip/hip_bf16.h>
#include <math.h>

typedef __attribute__((ext_vector_type(2))) float v2f;
typedef __attribute__((ext_vector_type(8))) float v8f;

#define WAVES_PER_BLOCK 8
#define XF_BLOCK 256

// ---------------------------------------------------------------------------
// WMMA helper: D = A(16x4 f32) * B(4x16 f32) + C(16x16 f32), wave32.
// 8-arg form: (neg_a, A, neg_b, B, c_mod, C, reuse_a, reuse_b)
// ---------------------------------------------------------------------------
__device__ __forceinline__ v8f wmma4(v2f a, v2f b, v8f c) {
  return __builtin_amdgcn_wmma_f32_16x16x4_f32(false, a, false, b, (short)0, c,
                                               false, false);
}

// ---------------------------------------------------------------------------
// Zero fill (vectorized b128 stores + scalar tail)
// ---------------------------------------------------------------------------
__global__ void zero4_kernel(float* __restrict__ p, int n) {
  int i = blockIdx.x * blockDim.x + threadIdx.x;
  int n4 = n >> 2;
  if (i < n4) ((float4*)p)[i] = make_float4(0.f, 0.f, 0.f, 0.f);
  int tail = (n4 << 2) + i;
  if (i < (n & 3)) p[tail] = 0.0f;
}

// ---------------------------------------------------------------------------
// Degree + normalization (improved GCN: deg = in_degree + 2)
// ---------------------------------------------------------------------------
__global__ void edge_deg_kernel(const int* __restrict__ ei,
                                float* __restrict__ deg, int E) {
  int e = blockIdx.x * blockDim.x + threadIdx.x;
  if (e < E) atomicAdd(&deg[ei[E + e]], 1.0f);
}

__global__ void node_norm_kernel(const float* __restrict__ deg,
                                 float* __restrict__ dinv,
                                 float* __restrict__ self_norm, int N) {
  int n = blockIdx.x * blockDim.x + threadIdx.x;
  if (n >= N) return;
  float di = rsqrtf(deg[n] + 2.0f);
  dinv[n] = di;
  self_norm[n] = 2.0f * di * di;
}

// ---------------------------------------------------------------------------
// WMMA transforms. One 16-row tile per wave.
// A layout: lanes 0-15 row m, VGPR0/1 = K,K+1 ; lanes 16-31 = K+2,K+3.
// B layout: lanes = column n, VGPR0/1 = rows K+2*lh, K+1+2*lh.
// D layout: VGPR v holds row (v + 8*lh), column = lane&15.
// Store guard is wave-uniform (full-tile fast path, straight-line stores).
// ---------------------------------------------------------------------------

// Layer 1: x [N,3] @ W1 [3,16]  (K padded 3->4 with zeros)
__global__ void xform_in_kernel(const float* __restrict__ x,
                                const float* __restrict__ W1,
                                float* __restrict__ out, int N, int tiles) {
  int tile = blockIdx.x * WAVES_PER_BLOCK + (threadIdx.x >> 5);
  if (tile >= tiles) return;  // wave-uniform exit
  int lane = threadIdx.x & 31;
  int lh = lane >> 4;
  int m = lane & 15;
  int rowBase = tile * 16;
  int r = rowBase + m;
  if (r > N - 1) r = N - 1;  // clamp loads, keep EXEC full
  const float* xr = x + (size_t)r * 3;

  // unconditional contiguous loads + cndmask selects (no divergent branch)
  float t0 = xr[0], t1 = xr[1], t2 = xr[2];
  float w0 = W1[0 * 16 + m], w1 = W1[1 * 16 + m], w2 = W1[2 * 16 + m];
  v2f a, b;
  a.x = lh ? t2 : t0;
  a.y = lh ? 0.0f : t1;
  b.x = lh ? w2 : w0;
  b.y = lh ? 0.0f : w1;
  v8f acc = {};
  acc = wmma4(a, b, acc);

  float* op = out + (size_t)rowBase * 16 + (size_t)lh * 128 + m;
  if (rowBase + 16 <= N) {  // uniform: straight-line stores
#pragma unroll
    for (int v = 0; v < 8; ++v) op[(size_t)v * 16] = acc[v];
  } else {
#pragma unroll
    for (int v = 0; v < 8; ++v)
      if (rowBase + v + 8 * lh < N) op[(size_t)v * 16] = acc[v];
  }
}

// Layer 2: h [N,16] @ W [16,16] — 4 chained 16x16x4 WMMAs
__global__ void xform_k16_kernel(const float* __restrict__ h,
                                 const float* __restrict__ W,
                                 float* __restrict__ out, int N, int tiles) {
  int tile = blockIdx.x * WAVES_PER_BLOCK + (threadIdx.x >> 5);
  if (tile >= tiles) return;
  int lane = threadIdx.x & 31;
  int lh = lane >> 4;
  int m = lane & 15;
  int rowBase = tile * 16;
  int r = rowBase + m;
  if (r > N - 1) r = N - 1;
  const float* hr = h + (size_t)r * 16;

  v8f acc = {};
#pragma unroll
  for (int k = 0; k < 4; ++k) {
    int kk = 4 * k + 2 * lh;
    v2f a, b;
    a.x = hr[kk];
    a.y = hr[kk + 1];
    b.x = W[kk * 16 + m];
    b.y = W[(kk + 1) * 16 + m];
    acc = wmma4(a, b, acc);
  }

  float* op = out + (size_t)rowBase * 16 + (size_t)lh * 128 + m;
  if (rowBase + 16 <= N) {
#pragma unroll
    for (int v = 0; v < 8; ++v) op[(size_t)v * 16] = acc[v];
  } else {
#pragma unroll
    for (int v = 0; v < 8; ++v)
      if (rowBase + v + 8 * lh < N) op[(size_t)v * 16] = acc[v];
  }
}

// Layer 3: h [N,16] @ W3 [16,1] — B padded to 16 cols, only col 0 valid
__global__ void xform_out_kernel(const float* __restrict__ h,
                                 const float* __restrict__ W3,
                                 float* __restrict__ cout, int N, int tiles) {
  int tile = blockIdx.x * WAVES_PER_BLOCK + (threadIdx.x >> 5);
  if (tile >= tiles) return;
  int lane = threadIdx.x & 31;
  int lh = lane >> 4;
  int m = lane & 15;
  int rowBase = tile * 16;
  int r = rowBase + m;
  if (r > N - 1) r = N - 1;
  const float* hr = h + (size_t)r * 16;

  v8f acc = {};
#pragma unroll
  for (int k = 0; k < 4; ++k) {
    int kk = 4 * k + 2 * lh;
    v2f a, b;
    a.x = hr[kk];
    a.y = hr[kk + 1];
    b.x = (m == 0) ? W3[kk] : 0.0f;
    b.y = (m == 0) ? W3[kk + 1] : 0.0f;
    acc = wmma4(a, b, acc);
  }

  if (m == 0) {
    float* op = cout + rowBase + 8 * lh;
    if (rowBase + 16 <= N) {
#pragma unroll
      for (int v = 0; v < 8; ++v) op[v] = acc[v];
    } else {
#pragma unroll
      for (int v = 0; v < 8; ++v)
        if (rowBase + v + 8 * lh < N) op[v] = acc[v];
    }
  }
}

// ---------------------------------------------------------------------------
// Edge aggregation: agg[col] += h_t[row] * dinv[row]*dinv[col]
// (agg buffer = 6.4 MB -> lives in L2; float atomics resolve there)
// ---------------------------------------------------------------------------
__global__ void agg16_kernel(const int* __restrict__ ei,
                             const float* __restrict__ dinv,
                             const float* __restrict__ ht,
                             float* __restrict__ agg, int E) {
  int e = blockIdx.x * blockDim.x + threadIdx.x;
  if (e >= E) return;
  int r = ei[e];
  int c = ei[E + e];
  float w = dinv[r] * dinv[c];
  const float4* hp = (const float4*)(ht + (size_t)r * 16);
  float4 h0 = hp[0], h1 = hp[1], h2 = hp[2], h3 = hp[3];
  float* ap = agg + (size_t)c * 16;
  atomicAdd(ap + 0, h0.x * w);  atomicAdd(ap + 1, h0.y * w);
  atomicAdd(ap + 2, h0.z * w);  atomicAdd(ap + 3, h0.w * w);
  atomicAdd(ap + 4, h1.x * w);  atomicAdd(ap + 5, h1.y * w);
  atomicAdd(ap + 6, h1.z * w);  atomicAdd(ap + 7, h1.w * w);
  atomicAdd(ap + 8, h2.x * w);  atomicAdd(ap + 9, h2.y * w);
  atomicAdd(ap + 10, h2.z * w); atomicAdd(ap + 11, h2.w * w);
  atomicAdd(ap + 12, h3.x * w); atomicAdd(ap + 13, h3.y * w);
  atomicAdd(ap + 14, h3.z * w); atomicAdd(ap + 15, h3.w * w);
}

__global__ void agg1_kernel(const int* __restrict__ ei,
                            const float* __restrict__ dinv,
                            const float* __restrict__ ct,
                            float* __restrict__ agg, int E) {
  int e = blockIdx.x * blockDim.x + threadIdx.x;
  if (e >= E) return;
  int r = ei[e];
  int c = ei[E + e];
  atomicAdd(&agg[c], ct[r] * dinv[r] * dinv[c]);
}

// ---------------------------------------------------------------------------
// Finalize: out = [relu](agg + self_norm*h_t + b)   (in-place on agg)
// ---------------------------------------------------------------------------
__global__ void finalize16_kernel(float* __restrict__ agg,
                                  const float* __restrict__ ht,
                                  const float* __restrict__ self_norm,
                                  const float* __restrict__ bias, int N,
                                  int do_relu) {
  int idx = blockIdx.x * blockDim.x + threadIdx.x;
  if (idx >= N * 16) return;
  int n = idx >> 4;
  int f = idx & 15;
  float v = agg[idx] + self_norm[n] * ht[idx] + bias[f];
  if (do_relu) v = fmaxf(v, 0.0f);
  agg[idx] = v;
}

__global__ void finalize1_kernel(float* __restrict__ ct,
                                 const float* __restrict__ aggc,
                                 const float* __restrict__ self_norm,
                                 const float* __restrict__ b3, int N) {
  int n = blockIdx.x * blockDim.x + threadIdx.x;
  if (n < N) ct[n] = aggc[n] + self_norm[n] * ct[n] + b3[0];
}

// ---------------------------------------------------------------------------
// Masked softmax: monotone uint encoding for atomicMax on floats
// ---------------------------------------------------------------------------
__device__ __forceinline__ unsigned fenc(float f) {
  unsigned u = __float_as_uint(f);
  return (u & 0x80000000u) ? ~u : (u | 0x80000000u);
}
__device__ __forceinline__ float fdec(unsigned u) {
  unsigned b = (u & 0x80000000u) ? (u ^ 0x80000000u) : ~u;
  return __uint_as_float(b);
}

__global__ void max_kernel(const float* __restrict__ c,
                           const unsigned char* __restrict__ mask,
                           unsigned* __restrict__ gmax, int N) {
  __shared__ unsigned smax;
  if (threadIdx.x == 0) smax = 0u;
  __syncthreads();
  int n = blockIdx.x * blockDim.x + threadIdx.x;
  if (n < N && mask[n]) atomicMax(&smax, fenc(c[n]));
  __syncthreads();
  if (threadIdx.x == 0) atomicMax(gmax, smax);
}

__global__ void expsum_kernel(const float* __restrict__ c,
                              const unsigned char* __restrict__ mask,
                              const unsigned* __restrict__ gmax,
                              float* __restrict__ gsum, int N) {
  __shared__ float ssum;
  if (threadIdx.x == 0) ssum = 0.0f;
  __syncthreads();
  int n = blockIdx.x * blockDim.x + threadIdx.x;
  float gm = fdec(*gmax);
  if (n < N && mask[n]) atomicAdd(&ssum, __expf(c[n] - gm));
  __syncthreads();
  if (threadIdx.x == 0) atomicAdd(gsum, ssum);
}

__global__ void softmax_write_kernel(const float* __restrict__ c,
                                     const unsigned char* __restrict__ mask,
                                     const unsigned* __restrict__ gmax,
                                     const float* __restrict__ gsum,
                                     float* __restrict__ out, int N) {
  int n = blockIdx.x * blockDim.x + threadIdx.x;
  if (n >= N) return;
  float gm = fdec(*gmax);
  float inv = 1.0f / (*gsum);
  out[n] = mask[n] ? __expf(c[n] - gm) * inv : 0.0f;
}

// ---------------------------------------------------------------------------
// Value head: mean-pool h2 then dot with fc_w
// ---------------------------------------------------------------------------
__global__ void pool_kernel(const float* __restrict__ h2,
                            float* __restrict__ vsum, int N) {
  __shared__ float sp[16];
  if (threadIdx.x < 16) sp[threadIdx.x] = 0.0f;
  __syncthreads();
  int n = blockIdx.x * blockDim.x + threadIdx.x;
  if (n < N) {
    const float* hr = h2 + (size_t)n * 16;
#pragma unroll
    for (int f = 0; f < 16; ++f) atomicAdd(&sp[f], hr[f]);
  }
  __syncthreads();
  if (threadIdx.x < 16) atomicAdd(&vsum[threadIdx.x], sp[threadIdx.x]);
}

__global__ void value_kernel(const float* __restrict__ vsum,
                             const float* __restrict__ fcw,
                             const float* __restrict__ fcb,
                             float* __restrict__ out_val, int N) {
  float invN = 1.0f / (float)N;
  float v = fcb[0];
#pragma unroll
  for (int f = 0; f < 16; ++f) v += vsum[f] * invN * fcw[f];
  out_val[0] = v;
}

// ---------------------------------------------------------------------------
// Host launcher
// ---------------------------------------------------------------------------
extern "C" void kernel_launch(void* const* d_in, const int* in_sizes, int n_in,
                              void* d_out, int out_size, void* d_ws,
                              size_t ws_size, hipStream_t stream) {
  const float* x = (const float*)d_in[0];
  const int* ei = (const int*)d_in[1];
  const unsigned char* choices = (const unsigned char*)d_in[2];
  const float* W1 = (const float*)d_in[3];
  const float* b1 = (const float*)d_in[4];
  const float* W2 = (const float*)d_in[5];
  const float* b2 = (const float*)d_in[6];
  const float* W3 = (const float*)d_in[7];
  const float* b3 = (const float*)d_in[8];
  const float* fcw = (const float*)d_in[9];
  const float* fcb = (const float*)d_in[10];

  const int N = in_sizes[0] / 3;
  const int E = in_sizes[1] / 2;
  float* out = (float*)d_out;

  // Workspace layout (floats): 37N + 32  (~14.8 MB for N=100k)
  float* ws = (float*)d_ws;
  size_t Ns = (size_t)N;
  float* dinv = ws;
  float* self_norm = ws + Ns;
  float* bufA = ws + 2 * Ns;   // 16N: transformed features h_t
  float* bufB = ws + 18 * Ns;  // 16N: aggregation / layer output
  float* c_t = ws + 34 * Ns;   // N
  float* aggc = ws + 35 * Ns;  // N
  float* deg = ws + 36 * Ns;   // N
  float* vsum = ws + 37 * Ns;  // 16
  float* gsum = vsum + 16;     // 1
  unsigned* gmax = (unsigned*)(gsum + 1);  // 1 (encoded float, 0 == -inf)

  const int tiles = (N + 15) / 16;
  const dim3 xfGrid((tiles + WAVES_PER_BLOCK - 1) / WAVES_PER_BLOCK);
  const int nbN = (N + 255) / 256;
  const int nbE = (E + 255) / 256;
  const int nbNF4 = (N * 4 + 255) / 256;  // N*16 floats / 4-wide stores
  const int nbN4 = (N / 4 + 256) / 256;

  // --- degrees & norms ---
  zero4_kernel<<<nbN4, 256, 0, stream>>>(deg, N);
  edge_deg_kernel<<<nbE, 256, 0, stream>>>(ei, deg, E);
  node_norm_kernel<<<nbN, 256, 0, stream>>>(deg, dinv, self_norm, N);

  // --- layer 1: x @ W1, aggregate, relu ---
  xform_in_kernel<<<xfGrid, XF_BLOCK, 0, stream>>>(x, W1, bufA, N, tiles);
  zero4_kernel<<<nbNF4, 256, 0, stream>>>(bufB, N * 16);
  agg16_kernel<<<nbE, 256, 0, stream>>>(ei, dinv, bufA, bufB, E);
  finalize16_kernel<<<(N * 16 + 255) / 256, 256, 0, stream>>>(bufB, bufA,
                                                              self_norm, b1, N, 1);

  // --- layer 2: h1 @ W2, aggregate, relu ---
  xform_k16_kernel<<<xfGrid, XF_BLOCK, 0, stream>>>(bufB, W2, bufA, N, tiles);
  zero4_kernel<<<nbNF4, 256, 0, stream>>>(bufB, N * 16);
  agg16_kernel<<<nbE, 256, 0, stream>>>(ei, dinv, bufA, bufB, E);
  finalize16_kernel<<<(N * 16 + 255) / 256, 256, 0, stream>>>(bufB, bufA,
                                                              self_norm, b2, N, 1);
  // bufB now holds h2 [N,16]

  // --- layer 3: h2 @ W3, aggregate (1 feature) ---
  xform_out_kernel<<<xfGrid, XF_BLOCK, 0, stream>>>(bufB, W3, c_t, N, tiles);
  zero4_kernel<<<nbN4, 256, 0, stream>>>(aggc, N);
  agg1_kernel<<<nbE, 256, 0, stream>>>(ei, dinv, c_t, aggc, E);
  finalize1_kernel<<<nbN, 256, 0, stream>>>(c_t, aggc, self_norm, b3, N);

  // --- heads: masked softmax + mean-pool value ---
  zero4_kernel<<<1, 32, 0, stream>>>(vsum, 18);  // vsum[16], gsum, gmax(enc 0 = -inf)
  max_kernel<<<nbN, 256, 0, stream>>>(c_t, choices, gmax, N);
  expsum_kernel<<<nbN, 256, 0, stream>>>(c_t, choices, gmax, gsum, N);
  pool_kernel<<<nbN, 256, 0, stream>>>(bufB, vsum, N);
  softmax_write_kernel<<<nbN, 256, 0, stream>>>(c_t, choices, gmax, gsum, out, N);
  value_kernel<<<1, 1, 0, stream>>>(vsum, fcw, fcb, out + N, N);
}